// MAE_75849122447787
// MI455X (gfx1250) — compile-verified
//
#include <hip/hip_runtime.h>
#include <hip/hip_bf16.h>

#define EMB   128          // embedding dim
#define KF    8            // K_FACTOR
#define TOPN  4
#define NPART 64           // partial copies of rel accumulator (contention spread)

typedef __attribute__((ext_vector_type(2))) float v2f;
typedef __attribute__((ext_vector_type(8))) float v8f;

// ---------------------------------------------------------------------------
// Per-relation stable descending argsort of rel_att row -> position table.
// perm[t*8+k] = p in [0,3] if block k is the p-th largest (top-4), else -1.
// Row t == num_rel is the zero pad row (self loops) -> top = [0,1,2,3].
// ---------------------------------------------------------------------------
__global__ void perm_kernel(const float* __restrict__ rel_att,
                            int* __restrict__ perm, int num_rel) {
    int t = blockIdx.x * blockDim.x + threadIdx.x;
    if (t > num_rel) return;
    float a[KF];
    for (int k = 0; k < KF; ++k)
        a[k] = (t < num_rel) ? rel_att[t * KF + k] : 0.0f;
    int posof[KF];
    bool used[KF];
    for (int k = 0; k < KF; ++k) { posof[k] = -1; used[k] = false; }
    for (int p = 0; p < TOPN; ++p) {              // stable selection (strict >)
        int best = 0; float bv = -3.402823466e38f;
        for (int j = 0; j < KF; ++j)
            if (!used[j] && a[j] > bv) { bv = a[j]; best = j; }
        used[best] = true; posof[best] = p;
    }
    for (int k = 0; k < KF; ++k) perm[t * KF + k] = posof[k];
}

// ---------------------------------------------------------------------------
// One wave32 per edge. lane covers 4 consecutive floats (float4 = b128 load).
// block k = lane>>2 (8 blocks of 16 floats), sub = lane&3.
//   ent_sum[dst] += h + (pos>=0 ? r[t, pos*16 ...] : 0)
//   rel_part[part][t, pos*16 ...] += (t_row - h) for top lanes
// Self-loop edges are NOT processed here (folded into finalize).
// ---------------------------------------------------------------------------
__global__ __launch_bounds__(256) void edge_kernel(
    const float* __restrict__ x, const float* __restrict__ r,
    const int* __restrict__ perm,
    const int* __restrict__ e_head, const int* __restrict__ e_tgt,
    const int* __restrict__ e_type,
    float* __restrict__ ent_sum, float* __restrict__ ent_cnt,
    float* __restrict__ rel_part, float* __restrict__ rel_cnt_part,
    int n_edges, int rel_p, int cnt_stride)
{
    int edge = (blockIdx.x * blockDim.x + threadIdx.x) >> 5;
    int lane = threadIdx.x & 31;
    if (edge >= n_edges) return;

    int src = e_head[edge];
    int dst = e_tgt[edge];
    int t   = e_type[edge];
    int k   = lane >> 2;
    int sub = lane & 3;
    int pos = perm[t * KF + k];

    const float4* x4 = (const float4*)x;
    float4 h = x4[(size_t)src * (EMB / 4) + lane];
    float4 v = h;
    int part = blockIdx.x & (NPART - 1);

    if (pos >= 0) {
        const float4* r4 = (const float4*)r;
        float4 rv = r4[(size_t)t * 16 + pos * 4 + sub];
        v.x += rv.x; v.y += rv.y; v.z += rv.z; v.w += rv.w;
        float4 tt = x4[(size_t)dst * (EMB / 4) + lane];
        float* rp = rel_part + (size_t)part * rel_p
                  + (size_t)t * 64 + pos * 16 + sub * 4;
        atomicAdd(rp + 0, tt.x - h.x);
        atomicAdd(rp + 1, tt.y - h.y);
        atomicAdd(rp + 2, tt.z - h.z);
        atomicAdd(rp + 3, tt.w - h.w);
    }
    float* es = ent_sum + (size_t)dst * EMB + lane * 4;
    atomicAdd(es + 0, v.x);
    atomicAdd(es + 1, v.y);
    atomicAdd(es + 2, v.z);
    atomicAdd(es + 3, v.w);
    if (lane == 0) {
        atomicAdd(ent_cnt + dst, 1.0f);
        atomicAdd(rel_cnt_part + part * cnt_stride + t, 1.0f);
    }
}

// Sum the 64 per-partition relation counts.
__global__ void rel_cnt_reduce(const float* __restrict__ rel_cnt_part,
                               float* __restrict__ tot, int num_rel, int cnt_stride) {
    int t = blockIdx.x * blockDim.x + threadIdx.x;
    if (t > num_rel) return;
    float s = 0.0f;
    for (int p = 0; p < NPART; ++p) s += rel_cnt_part[p * cnt_stride + t];
    tot[t] = s;
}

// ---------------------------------------------------------------------------
// Reduce the 64 rel partials with V_WMMA_F32_16X16X4_F32.
// A = ones(16x4)  =>  D[m][n] = sum_k B[k][n] + C[m][n]: column sums.
// Each wave reduces 16 flat positions; 16 chained WMMAs consume all 64 parts.
// Every lane%16 group loads the SAME position, so column n = base+n regardless
// of the exact K striping, and all D rows are identical -> read c[0].
// ---------------------------------------------------------------------------
__global__ __launch_bounds__(256) void rel_reduce_wmma(
    const float* __restrict__ rel_part, const float* __restrict__ rel_cnt_tot,
    float* __restrict__ r_out, int num_rel, int rel_p)
{
    int wave = (blockIdx.x * blockDim.x + threadIdx.x) >> 5;
    int lane = threadIdx.x & 31;
    int base = wave * 16;
    if (base >= rel_p) return;            // wave-uniform: EXEC stays all-1s

    int half = lane >> 4;
    int pos  = base + (lane & 15);

    v8f c = {0.f, 0.f, 0.f, 0.f, 0.f, 0.f, 0.f, 0.f};
    v2f a; a[0] = 1.0f; a[1] = 1.0f;
    for (int i = 0; i < NPART / 4; ++i) {
        int p0 = 4 * i + 2 * half;        // this lane supplies partials p0, p0+1
        v2f b;
        b[0] = rel_part[(size_t)p0 * rel_p + pos];
        b[1] = rel_part[(size_t)(p0 + 1) * rel_p + pos];
        c = __builtin_amdgcn_wmma_f32_16x16x4_f32(
                false, a, false, b, (short)0, c, false, false);
    }
    if (half == 0) {                      // one writer per position
        int t = pos >> 6;
        if (t < num_rel) {                // drop the pad relation row
            float cnt = rel_cnt_tot[t];
            r_out[pos] = (cnt > 0.0f) ? (c[0] / cnt) : 0.0f;
        }
    }
}

// x_out[i] = relu( (ent_sum[i] + x_in[i]) / (cnt + 1) )   (self loop folded in)
__global__ __launch_bounds__(256) void ent_finalize(
    const float* __restrict__ ent_sum, const float* __restrict__ x_in,
    const float* __restrict__ ent_cnt, float* __restrict__ x_out, int num_ent)
{
    int i = blockIdx.x * blockDim.x + threadIdx.x;   // over num_ent * 32 float4s
    if (i >= num_ent * (EMB / 4)) return;
    int node = i >> 5;
    const float4* s4 = (const float4*)ent_sum;
    const float4* x4 = (const float4*)x_in;
    float4 s = s4[i];
    float4 xv = x4[i];
    float inv = 1.0f / (ent_cnt[node] + 1.0f);
    float4 o;
    o.x = fmaxf((s.x + xv.x) * inv, 0.0f);
    o.y = fmaxf((s.y + xv.y) * inv, 0.0f);
    o.z = fmaxf((s.z + xv.z) * inv, 0.0f);
    o.w = fmaxf((s.w + xv.w) * inv, 0.0f);
    ((float4*)x_out)[i] = o;
}

extern "C" void kernel_launch(void* const* d_in, const int* in_sizes, int n_in,
                              void* d_out, int out_size, void* d_ws, size_t ws_size,
                              hipStream_t stream)
{
    (void)n_in; (void)out_size; (void)ws_size;
    const float* x0   = (const float*)d_in[0];   // ent_embeddings (num_ent,128)
    const float* r0   = (const float*)d_in[1];   // rel_embeddings (num_rel,64)
    const float* ratt = (const float*)d_in[2];   // rel_att        (num_rel,8)
    const int*   eidx = (const int*)d_in[3];     // edge_index     (2,E)
    const int*   etyp = (const int*)d_in[4];     // edge_type      (E,)

    const int num_ent = in_sizes[0] / EMB;
    const int num_rel = in_sizes[2] / KF;
    const int n_edges = in_sizes[4];
    const int rel_p   = (num_rel + 1) * 64;                // flat rel positions
    const int cstride = (num_rel + 1 + 63) & ~63;          // padded count stride

    const int* e_head = eidx;                // edge_index[0]
    const int* e_tgt  = eidx + n_edges;      // edge_index[1]

    // ---- carve workspace ----
    char* w = (char*)d_ws;
    auto carve = [&](size_t bytes) -> char* {
        char* p = w; w += (bytes + 255) & ~(size_t)255; return p;
    };
    float* ent_sum      = (float*)carve((size_t)num_ent * EMB * 4);
    float* x1           = (float*)carve((size_t)num_ent * EMB * 4);
    float* rel_part     = (float*)carve((size_t)NPART * rel_p * 4);
    float* rel_cnt_part = (float*)carve((size_t)NPART * cstride * 4);
    float* rel_cnt_tot  = (float*)carve((size_t)cstride * 4);
    float* r1           = (float*)carve((size_t)rel_p * 4);
    float* ent_cnt      = (float*)carve((size_t)num_ent * 4);
    int*   perm         = (int*)carve((size_t)(num_rel + 1) * KF * 4);

    float* x_out = (float*)d_out;
    float* r_out = x_out + (size_t)num_ent * EMB;

    // ---- per-relation top-4 table (rel_att is layer-invariant) ----
    perm_kernel<<<(num_rel + 256) / 256, 256, 0, stream>>>(ratt, perm, num_rel);

    const int edge_blocks = (n_edges + 7) / 8;                  // 8 waves/block
    const int waves_r     = rel_p / 16;
    const int rr_blocks   = (waves_r + 7) / 8;
    const int fin_blocks  = (num_ent * (EMB / 4) + 255) / 256;
    const int cnt_blocks  = (num_rel + 256) / 256;

    const float* xin = x0;
    const float* rin = r0;
    for (int layer = 0; layer < 2; ++layer) {
        float* xo = (layer == 0) ? x1 : x_out;
        float* ro = (layer == 0) ? r1 : r_out;

        hipMemsetAsync(ent_sum,      0, (size_t)num_ent * EMB * 4,   stream);
        hipMemsetAsync(ent_cnt,      0, (size_t)num_ent * 4,         stream);
        hipMemsetAsync(rel_part,     0, (size_t)NPART * rel_p * 4,   stream);
        hipMemsetAsync(rel_cnt_part, 0, (size_t)NPART * cstride * 4, stream);

        edge_kernel<<<edge_blocks, 256, 0, stream>>>(
            xin, rin, perm, e_head, e_tgt, etyp,
            ent_sum, ent_cnt, rel_part, rel_cnt_part,
            n_edges, rel_p, cstride);

        rel_cnt_reduce<<<cnt_blocks, 256, 0, stream>>>(
            rel_cnt_part, rel_cnt_tot, num_rel, cstride);

        rel_reduce_wmma<<<rr_blocks, 256, 0, stream>>>(
            rel_part, rel_cnt_tot, ro, num_rel, rel_p);

        ent_finalize<<<fin_blocks, 256, 0, stream>>>(
            ent_sum, xin, ent_cnt, xo, num_ent);

        xin = xo; rin = ro;
    }
}